// OptionSelector_57561151701695
// MI455X (gfx1250) — compile-verified
//
#include <hip/hip_runtime.h>

// ---------------------------------------------------------------------------
// Problem dimensions (fixed by the reference)
// ---------------------------------------------------------------------------
#define B_   256
#define T_   2000
#define S_   39
#define SP_  40                  // S_ padded to multiple of 4 (zero-filled)
#define L_   768
#define H_   128
#define D_   128
#define CD_  16
#define K_   20
#define HOR  10
#define TS_  200                 // T_/HOR
#define M1_  (B_ * T_)           // 512000 rows of state_emb_full
#define M2_  (B_ * TS_)          // 51200 rows of the option path

// Output layout (flat f32, reference return order)
#define OFF_OPT    0
#define N_OPT      (M2_ * D_)              // 6,553,600
#define OFF_IDX    (OFF_OPT + N_OPT)       // 6,553,600
#define N_IDX      (M2_)                   // 51,200
#define OFF_COMMIT (OFF_IDX + N_IDX)       // 6,604,800
#define OFF_SENT   (OFF_COMMIT + 1)
#define OFF_BENT   (OFF_SENT + 1)
#define OFF_SE     (OFF_BENT + 1)          // 6,604,803 (odd -> B32 stores only)

typedef __attribute__((ext_vector_type(2))) float v2f;
typedef __attribute__((ext_vector_type(8))) float v8f;

// D = A(16x4,f32) * B(4x16,f32) + C(16x16,f32)  -- CDNA5 V_WMMA_F32_16X16X4_F32
__device__ __forceinline__ v8f wmma4(v2f a, v2f b, v8f c) {
  return __builtin_amdgcn_wmma_f32_16x16x4_f32(
      /*neg_a=*/false, a, /*neg_b=*/false, b,
      /*c_mod=*/(short)0, c, /*reuse_a=*/false, /*reuse_b=*/false);
}

__device__ __forceinline__ float wave_red_sum(float v) {
  #pragma unroll
  for (int off = 16; off > 0; off >>= 1) v += __shfl_xor(v, off, 32);
  return v;
}

// out[16][NOUT] = in[16][KIN] @ W[KIN][NOUT] + bias[NOUT]
template <int KIN, int NOUT>
__device__ __forceinline__ void gemm_tile(const float* in, const float* __restrict__ W,
                                          const float* bias, float* out, size_t ostride,
                                          int lane) {
  const int l = lane & 15, hi = lane >> 4;
  #pragma unroll
  for (int nt = 0; nt < NOUT / 16; ++nt) {
    v8f acc = {0.f, 0.f, 0.f, 0.f, 0.f, 0.f, 0.f, 0.f};
    #pragma unroll
    for (int kc = 0; kc < KIN / 4; ++kc) {
      const int ka = kc * 4 + hi * 2;
      v2f a, b;
      a.x = in[l * KIN + ka];
      a.y = in[l * KIN + ka + 1];
      b.x = W[ka * NOUT + nt * 16 + l];
      b.y = W[(ka + 1) * NOUT + nt * 16 + l];
      acc = wmma4(a, b, acc);
    }
    const int col = nt * 16 + l;
    const float bv = bias[col];
    #pragma unroll
    for (int r = 0; r < 8; ++r)
      out[(size_t)(r + hi * 8) * ostride + col] = acc[r] + bv;
  }
}

// ---------------------------------------------------------------------------
// Kernel 0: zero scalar accumulators (commit, sample_entropy, avg_p[20])
// ---------------------------------------------------------------------------
__global__ void k_zero(float* accs) {
  if (threadIdx.x < 32) accs[threadIdx.x] = 0.f;
}

// ---------------------------------------------------------------------------
// Kernel 1: state_emb_full = states @ W_state + b_state   (bandwidth bound)
// 8 waves/block; block stages its 128 contiguous `states` rows via B128 loads
// into a K-padded LDS tile, WMMAs from LDS, and flushes C through LDS with
// lane-contiguous B32 stores (output base is odd -> cannot use B128).
// ---------------------------------------------------------------------------
__global__ __launch_bounds__(256) void k_state_emb(const float* __restrict__ states,
                                                   const float* __restrict__ W_state,
                                                   const float* __restrict__ b_state,
                                                   float* __restrict__ out) {
  __shared__ __align__(16) float sW[SP_ * H_];     // K-padded weights (row 39 = 0)
  __shared__ float sB[H_];
  __shared__ __align__(16) float sA[128 * SP_];    // 128 rows x 40 (col 39 = 0)
  __shared__ __align__(16) float sOut[8][16 * H_]; // per-wave C staging

  // stage weights (pad row 39 with zeros) + bias
  for (int i = threadIdx.x; i < S_ * H_; i += 256) sW[i] = W_state[i];
  for (int i = threadIdx.x; i < H_; i += 256) {
    sW[S_ * H_ + i] = 0.f;
    sB[i] = b_state[i];
  }

  // stage the block's 128x39 states tile: contiguous 4992 floats -> B128 loads
  const size_t blk_row0 = (size_t)blockIdx.x * 128;
  const float4* src = (const float4*)(states + blk_row0 * S_);
  for (int i = threadIdx.x; i < (128 * S_) / 4; i += 256) {
    const float4 v = src[i];
    const float vv[4] = {v.x, v.y, v.z, v.w};
    #pragma unroll
    for (int j = 0; j < 4; ++j) {
      const int idx = i * 4 + j;
      const int r = idx / S_;
      const int c = idx - r * S_;
      sA[r * SP_ + c] = vv[j];
    }
  }
  for (int i = threadIdx.x; i < 128; i += 256) sA[i * SP_ + S_] = 0.f;
  __syncthreads();

  const int wave = threadIdx.x >> 5;
  const int lane = threadIdx.x & 31;
  const int l = lane & 15, hi = lane >> 4;
  const float* arow = &sA[(wave * 16 + l) * SP_];

  v8f acc[8];
  #pragma unroll
  for (int nt = 0; nt < 8; ++nt)
    acc[nt] = (v8f){0.f, 0.f, 0.f, 0.f, 0.f, 0.f, 0.f, 0.f};

  #pragma unroll
  for (int kc = 0; kc < SP_ / 4; ++kc) {
    const int ka = kc * 4 + hi * 2;
    v2f a;
    a.x = arow[ka];
    a.y = arow[ka + 1];
    #pragma unroll
    for (int nt = 0; nt < 8; ++nt) {
      v2f b;
      b.x = sW[ka * H_ + nt * 16 + l];
      b.y = sW[(ka + 1) * H_ + nt * 16 + l];
      acc[nt] = wmma4(a, b, acc[nt]);
    }
  }

  // C fragments -> per-wave LDS tile (+bias)
  float* ot = sOut[wave];
  #pragma unroll
  for (int nt = 0; nt < 8; ++nt) {
    const int col = nt * 16 + l;
    const float bv = sB[col];
    #pragma unroll
    for (int r = 0; r < 8; ++r)
      ot[(r + hi * 8) * H_ + col] = acc[nt][r] + bv;
  }
  // lane-contiguous coalesced stores (128B per instruction across the wave)
  float* dst = out + (blk_row0 + (size_t)wave * 16) * H_;
  #pragma unroll
  for (int i = 0; i < (16 * H_) / 32; ++i)
    dst[i * 32 + lane] = ot[i * 32 + lane];
}

// ---------------------------------------------------------------------------
// Kernel 2: le_all = word_embeddings @ W_lang + b_lang   (256x768 @ 768x128)
// ---------------------------------------------------------------------------
__global__ __launch_bounds__(256) void k_lang_emb(const float* __restrict__ we,
                                                  const float* __restrict__ W_lang,
                                                  const float* __restrict__ b_lang,
                                                  float* __restrict__ le_out) {
  const int wave = threadIdx.x >> 5;
  const int lane = threadIdx.x & 31;
  const int l = lane & 15, hi = lane >> 4;
  const int row0 = (blockIdx.x * 8 + wave) * 16;
  const float* arow = we + (size_t)(row0 + l) * L_;

  v8f acc[8];
  #pragma unroll
  for (int nt = 0; nt < 8; ++nt)
    acc[nt] = (v8f){0.f, 0.f, 0.f, 0.f, 0.f, 0.f, 0.f, 0.f};

  for (int kc = 0; kc < L_ / 4; ++kc) {
    const int ka = kc * 4 + hi * 2;
    if (ka + 32 < L_)  // stream W_lang (393KB) ahead of use
      __builtin_prefetch(&W_lang[(size_t)(ka + 32) * H_ + lane * 4], 0, 3);
    v2f a;
    a.x = arow[ka];
    a.y = arow[ka + 1];
    #pragma unroll
    for (int nt = 0; nt < 8; ++nt) {
      v2f b;
      b.x = W_lang[ka * H_ + nt * 16 + l];
      b.y = W_lang[(ka + 1) * H_ + nt * 16 + l];
      acc[nt] = wmma4(a, b, acc[nt]);
    }
  }
  #pragma unroll
  for (int nt = 0; nt < 8; ++nt) {
    const int col = nt * 16 + l;
    const float bv = b_lang[col];
    #pragma unroll
    for (int r = 0; r < 8; ++r)
      le_out[(size_t)(row0 + r + hi * 8) * H_ + col] = acc[nt][r] + bv;
  }
}

// ---------------------------------------------------------------------------
// Kernel 3: fused option path. 4 waves/block, 16 rows per wave.
// inp=[le,se] -> W0 -> W1 -> W2 -> W_pin -> VQ/argmin/entropy -> q @ W_pout
// ---------------------------------------------------------------------------
#define WAVES_ 4
__global__ __launch_bounds__(32 * WAVES_) void k_option(
    const float* __restrict__ states, const float* __restrict__ le_all,
    const float* __restrict__ W0, const float* __restrict__ b0,
    const float* __restrict__ W1, const float* __restrict__ b1,
    const float* __restrict__ W2, const float* __restrict__ b2,
    const float* __restrict__ W_state, const float* __restrict__ b_state,
    const float* __restrict__ W_pin, const float* __restrict__ b_pin,
    const float* __restrict__ W_pout, const float* __restrict__ b_pout,
    const float* __restrict__ cb, float* accs,
    float* __restrict__ out_opt, float* __restrict__ out_idx) {
  __shared__ float sW_state[S_ * H_];
  __shared__ float sb_state[H_];
  __shared__ float sW_pin[D_ * CD_];
  __shared__ float sb_pin[CD_];
  __shared__ float sW_pout[CD_ * D_];
  __shared__ float sb_pout[D_];
  __shared__ float scb[K_ * CD_];
  __shared__ __align__(16) float bufA[WAVES_][16 * 256];  // inp / h1 / x / q
  __shared__ __align__(16) float bufB[WAVES_][16 * 128];  // h0 / op / options

  const int nthr = 32 * WAVES_;
  for (int i = threadIdx.x; i < S_ * H_; i += nthr) sW_state[i] = W_state[i];
  for (int i = threadIdx.x; i < H_; i += nthr) sb_state[i] = b_state[i];
  for (int i = threadIdx.x; i < D_ * CD_; i += nthr) sW_pin[i] = W_pin[i];
  if (threadIdx.x < CD_) sb_pin[threadIdx.x] = b_pin[threadIdx.x];
  for (int i = threadIdx.x; i < CD_ * D_; i += nthr) sW_pout[i] = W_pout[i];
  for (int i = threadIdx.x; i < D_; i += nthr) sb_pout[i] = b_pout[i];
  for (int i = threadIdx.x; i < K_ * CD_; i += nthr) scb[i] = cb[i];
  __syncthreads();

  const int wave = threadIdx.x >> 5;
  const int lane = threadIdx.x & 31;
  const int l = lane & 15, hi = lane >> 4;
  const int row0 = (blockIdx.x * WAVES_ + wave) * 16;

  float* inp = bufA[wave];   // 16 x 256
  float* buf = bufB[wave];   // 16 x 128

  // --- build inp[:, 0:128] = le_all[batch(row)] ---------------------------
  for (int i = lane; i < 16 * H_; i += 32) {
    const int r = i >> 7, c = i & 127;
    const int gb = (row0 + r) / TS_;
    inp[r * 256 + c] = le_all[(size_t)gb * H_ + c];
  }

  // --- inp[:, 128:256] = se = states[b, t*HOR] @ W_state + b_state --------
  {
    const int gr = row0 + l;
    const int gb = gr / TS_;
    const int gt = (gr % TS_) * HOR;
    const float* arow = states + ((size_t)gb * T_ + gt) * S_;
    #pragma unroll
    for (int nt = 0; nt < 8; ++nt) {
      v8f acc = {0.f, 0.f, 0.f, 0.f, 0.f, 0.f, 0.f, 0.f};
      #pragma unroll
      for (int kc = 0; kc < 10; ++kc) {
        const int ka = kc * 4 + hi * 2;
        v2f a, b;
        a.x = (ka < S_) ? arow[ka] : 0.f;
        a.y = (ka + 1 < S_) ? arow[ka + 1] : 0.f;
        b.x = (ka < S_) ? sW_state[ka * H_ + nt * 16 + l] : 0.f;
        b.y = (ka + 1 < S_) ? sW_state[(ka + 1) * H_ + nt * 16 + l] : 0.f;
        acc = wmma4(a, b, acc);
      }
      const int col = nt * 16 + l;
      const float bv = sb_state[col];
      #pragma unroll
      for (int r = 0; r < 8; ++r)
        inp[(r + hi * 8) * 256 + 128 + col] = acc[r] + bv;
    }
  }

  // --- linear chain (weights stay in L2; activations ping-pong in LDS) ---
  gemm_tile<256, 128>(inp, W0, b0, buf, 128, lane);        // h0
  gemm_tile<128, 128>(buf, W1, b1, inp, 128, lane);        // h1 (reuse bufA)
  gemm_tile<128, 128>(inp, W2, b2, buf, 128, lane);        // option_preds
  gemm_tile<128, 16>(buf, sW_pin, sb_pin, inp, 16, lane);  // x (16x16 tile)

  // --- per-row VQ: d2, argmin, commitment, softmax entropies --------------
  float commit = 0.f, ent = 0.f;
  float p[K_];
  if (hi == 0) {
    float xv[CD_];
    #pragma unroll
    for (int c = 0; c < CD_; ++c) xv[c] = inp[l * CD_ + c];
    float xx = 0.f;
    #pragma unroll
    for (int c = 0; c < CD_; ++c) xx += xv[c] * xv[c];

    float d2a[K_];
    float dmin = 3.0e38f;
    int best = 0;
    for (int k = 0; k < K_; ++k) {
      float dot = 0.f, cc = 0.f;
      #pragma unroll
      for (int c = 0; c < CD_; ++c) {
        const float cv = scb[k * CD_ + c];
        dot += xv[c] * cv;
        cc += cv * cv;
      }
      const float d = xx - 2.f * dot + cc;
      d2a[k] = d;
      if (d < dmin) { dmin = d; best = k; }
    }
    out_idx[row0 + l] = (float)best;

    #pragma unroll
    for (int c = 0; c < CD_; ++c) {
      const float df = scb[best * CD_ + c] - xv[c];
      commit += df * df;
    }
    // log_softmax(-d2): logp_k = (dmin - d2_k) - log(sum exp(dmin - d2_j))
    float ssum = 0.f;
    for (int k = 0; k < K_; ++k) {
      const float e = __expf(dmin - d2a[k]);
      p[k] = e;
      ssum += e;
    }
    const float inv = 1.f / ssum;
    const float lse = __logf(ssum);
    for (int k = 0; k < K_; ++k) {
      const float pk = p[k] * inv;
      const float logpk = (dmin - d2a[k]) - lse;
      p[k] = pk;
      ent -= pk * logpk;
    }
    // q tile replaces x tile (consumed above)
    #pragma unroll
    for (int c = 0; c < CD_; ++c) inp[l * CD_ + c] = scb[best * CD_ + c];
  } else {
    #pragma unroll
    for (int k = 0; k < K_; ++k) p[k] = 0.f;
  }

  // wave reductions + global accumulation
  commit = wave_red_sum(commit);
  ent = wave_red_sum(ent);
  if (lane == 0) {
    atomicAdd(&accs[0], commit);
    atomicAdd(&accs[1], ent);
  }
  #pragma unroll
  for (int k = 0; k < K_; ++k) {
    const float pk = wave_red_sum(p[k]);
    if (lane == 0) atomicAdd(&accs[2 + k], pk);
  }

  // --- options = q @ W_pout + b_pout -> LDS, then coalesced B128 stores ---
  gemm_tile<16, 128>(inp, sW_pout, sb_pout, buf, 128, lane);
  float4* dst = (float4*)(out_opt + (size_t)row0 * D_);
  const float4* src4 = (const float4*)buf;
  #pragma unroll
  for (int i = 0; i < (16 * D_) / (4 * 32); ++i)
    dst[i * 32 + lane] = src4[i * 32 + lane];
}

// ---------------------------------------------------------------------------
// Kernel 4: finalize the three scalar outputs
// ---------------------------------------------------------------------------
__global__ void k_finalize(const float* accs, float* out_commit, float* out_sent,
                           float* out_bent) {
  if (threadIdx.x == 0) {
    *out_commit = 0.25f * accs[0] / (float)(M2_ * CD_);
    *out_sent = accs[1] / (float)M2_;
    float be = 0.f;
    for (int k = 0; k < K_; ++k) {
      const float ap = accs[2 + k] / (float)M2_;
      be -= ap * __logf(ap + 1e-10f);
    }
    *out_bent = be;
  }
}

// ---------------------------------------------------------------------------
extern "C" void kernel_launch(void* const* d_in, const int* in_sizes, int n_in,
                              void* d_out, int out_size, void* d_ws, size_t ws_size,
                              hipStream_t stream) {
  const float* we      = (const float*)d_in[0];
  const float* states  = (const float*)d_in[1];
  const float* W_state = (const float*)d_in[2];
  const float* b_state = (const float*)d_in[3];
  const float* W_lang  = (const float*)d_in[4];
  const float* b_lang  = (const float*)d_in[5];
  const float* W0      = (const float*)d_in[6];
  const float* b0      = (const float*)d_in[7];
  const float* W1      = (const float*)d_in[8];
  const float* b1      = (const float*)d_in[9];
  const float* W2      = (const float*)d_in[10];
  const float* b2      = (const float*)d_in[11];
  const float* W_pin   = (const float*)d_in[12];
  const float* b_pin   = (const float*)d_in[13];
  const float* W_pout  = (const float*)d_in[14];
  const float* b_pout  = (const float*)d_in[15];
  const float* cb      = (const float*)d_in[16];

  float* out = (float*)d_out;
  float* le_all = (float*)d_ws;               // 256*128 f32
  float* accs = le_all + (size_t)B_ * H_;     // 22 f32 accumulators

  k_zero<<<1, 32, 0, stream>>>(accs);
  k_state_emb<<<M1_ / 128, 256, 0, stream>>>(states, W_state, b_state,
                                             out + OFF_SE);
  k_lang_emb<<<B_ / (16 * 8), 256, 0, stream>>>(we, W_lang, b_lang, le_all);
  k_option<<<M2_ / (16 * WAVES_), 32 * WAVES_, 0, stream>>>(
      states, le_all, W0, b0, W1, b1, W2, b2, W_state, b_state, W_pin, b_pin,
      W_pout, b_pout, cb, accs, out + OFF_OPT, out + OFF_IDX);
  k_finalize<<<1, 1, 0, stream>>>(accs, out + OFF_COMMIT, out + OFF_SENT,
                                  out + OFF_BENT);
}